// LQRActionNet_2697239462038
// MI455X (gfx1250) — compile-verified
//
#include <hip/hip_runtime.h>
#include <math.h>

#define HORIZON    5
#define DT         0.1f
#define X_REF_MAX  1.0f
#define Y_REF_MAX  0.8f
#define THETA_REF_MAX 1.5707963267948966f
#define V_MAXC     0.22f
#define W_MAXC     2.8f
#define EPS        0.0001f

#define BATCH      32768
#define LATENT     1024
#define ROWS_PER_WAVE   32
#define WAVES_PER_BLOCK 8
#define ROWS_PER_BLOCK  (ROWS_PER_WAVE * WAVES_PER_BLOCK)
#define KCH        (LATENT / 2)   // 512 K-pair chunks
#define PADN       5              // B columns stored in LDS

typedef __attribute__((ext_vector_type(2))) float v2f;
typedef __attribute__((ext_vector_type(8))) float v8f;

__device__ __forceinline__ float softplus_f(float x) {
    // numerically stable: max(x,0) + log1p(exp(-|x|))
    return fmaxf(x, 0.0f) + log1pf(expf(-fabsf(x)));
}

__global__ __launch_bounds__(256) void lqr_action_net_kernel(
    const float* __restrict__ latent,   // [BATCH][LATENT]
    const float* __restrict__ W,        // [5][LATENT]
    const float* __restrict__ bias,     // [5]
    const float* __restrict__ q_raw,    // [3]
    const float* __restrict__ r_raw,    // [2]
    const float* __restrict__ qf_raw,   // [3]
    float* __restrict__ out)            // [BATCH][2]
{
    // W staged as K-pair-interleaved float2: ldsW[c*PADN + n] = {W[n][2c], W[n][2c+1]}
    __shared__ v2f   ldsW[KCH * PADN];                              // 20 KB
    __shared__ float lds_raw[WAVES_PER_BLOCK][ROWS_PER_WAVE][8];    // 8 KB

    const int tid = threadIdx.x;

    // ---- preload W into LDS (B-fragment friendly layout) ----
    for (int idx = tid; idx < KCH * PADN; idx += 256) {
        const int c = idx / PADN;
        const int n = idx - c * PADN;
        const float* src = W + n * LATENT + 2 * c;
        v2f val;
        val.x = src[0];
        val.y = src[1];
        ldsW[idx] = val;
    }
    __syncthreads();

    const int wv   = tid >> 5;       // wave in block
    const int lane = tid & 31;
    const int n16  = lane & 15;      // N (B/C column) or M (A row) within tile
    const int hi   = lane >> 4;      // lane-half -> K offset {0,2}
    const int koff = hi * 2;

    const int waveBase = blockIdx.x * ROWS_PER_BLOCK + wv * ROWS_PER_WAVE;

    // A-fragment row streams (two 16-row tiles per wave)
    const float* rp0 = latent + (size_t)(waveBase + n16) * LATENT + koff;
    const float* rp1 = rp0 + (size_t)16 * LATENT;
    // B column for this lane: columns >=5 of C are garbage and discarded,
    // so just alias a valid column (n % 5) — WMMA columns are independent.
    const v2f* bp = ldsW + (n16 % PADN);

    v8f acc0 = {};
    v8f acc1 = {};

    #pragma unroll 4
    for (int k = 0; k < LATENT; k += 4) {
        v2f a0 = *(const v2f*)(rp0 + k);
        v2f a1 = *(const v2f*)(rp1 + k);
        v2f b  = bp[((k >> 1) + hi) * PADN];
        acc0 = __builtin_amdgcn_wmma_f32_16x16x4_f32(
            false, a0, false, b, (short)0, acc0, false, false);
        acc1 = __builtin_amdgcn_wmma_f32_16x16x4_f32(
            false, a1, false, b, (short)0, acc1, false, false);
    }

    // ---- scatter raw (cols 0..4) so each lane owns one full row ----
    if (n16 < 5) {
        #pragma unroll
        for (int j = 0; j < 8; ++j) {
            // C layout: VGPR j -> M=j (lanes 0-15) / M=8+j (lanes 16-31), N=lane&15
            lds_raw[wv][hi * 8 + j][n16]      = acc0[j];
            lds_raw[wv][16 + hi * 8 + j][n16] = acc1[j];
        }
    }
    __syncthreads();

    // ---- per-row scalar pipeline: one row per lane ----
    const int row = waveBase + lane;
    if (row >= BATCH) return;

    const float r0v = lds_raw[wv][lane][0] + bias[0];
    const float r1v = lds_raw[wv][lane][1] + bias[1];
    const float r2v = lds_raw[wv][lane][2] + bias[2];
    const float r3v = lds_raw[wv][lane][3] + bias[3];
    const float r4v = lds_raw[wv][lane][4] + bias[4];

    const float x  = tanhf(r0v) * X_REF_MAX;
    const float y  = tanhf(r1v) * Y_REF_MAX;
    const float th = tanhf(r2v) * THETA_REF_MAX;
    const float v  = tanhf(r3v) * V_MAXC;
    const float om = tanhf(r4v) * W_MAXC;

    const float ct = cosf(th);
    const float st = sinf(th);
    const float a02 = -DT * v * st;   // A[0][2]
    const float a12 =  DT * v * ct;   // A[1][2]

    const float q0 = softplus_f(q_raw[0]) + EPS;
    const float q1 = softplus_f(q_raw[1]) + EPS;
    const float q2 = softplus_f(q_raw[2]) + EPS;
    const float rr0 = softplus_f(r_raw[0]) + EPS;
    const float rr1 = softplus_f(r_raw[1]) + EPS;

    float p00 = softplus_f(qf_raw[0]) + EPS;
    float p11 = softplus_f(qf_raw[1]) + EPS;
    float p22 = softplus_f(qf_raw[2]) + EPS;
    float p01 = 0.0f, p02 = 0.0f, p12 = 0.0f;

    float k00 = 0.f, k01 = 0.f, k02 = 0.f, k10 = 0.f, k11 = 0.f, k12 = 0.f;

    #pragma unroll
    for (int it = 0; it < HORIZON; ++it) {
        // BtP = B^T P  (2x3), B col0 = DT*(ct,st,0), col1 = DT*(0,0,1)
        const float t00 = DT * (ct * p00 + st * p01);
        const float t01 = DT * (ct * p01 + st * p11);
        const float t02 = DT * (ct * p02 + st * p12);
        const float t10 = DT * p02;
        const float t11 = DT * p12;
        const float t12 = DT * p22;
        // S = R + BtP @ B
        const float S00 = rr0 + DT * (t00 * ct + t01 * st);
        const float S01 = DT * t02;
        const float S10 = DT * (t10 * ct + t11 * st);
        const float S11 = rr1 + DT * t12;
        // M = BtP @ A  (A = I with A02=a02, A12=a12)
        const float m00 = t00, m01 = t01;
        const float m02 = t00 * a02 + t01 * a12 + t02;
        const float m10 = t10, m11 = t11;
        const float m12 = t10 * a02 + t11 * a12 + t12;
        // K = S^-1 @ M
        const float inv = 1.0f / (S00 * S11 - S01 * S10);
        k00 = (S11 * m00 - S01 * m10) * inv;
        k01 = (S11 * m01 - S01 * m11) * inv;
        k02 = (S11 * m02 - S01 * m12) * inv;
        k10 = (S00 * m10 - S10 * m00) * inv;
        k11 = (S00 * m11 - S10 * m01) * inv;
        k12 = (S00 * m12 - S10 * m02) * inv;
        if (it == HORIZON - 1) break;  // only K from the last step is consumed
        // A^T P A (symmetric closed form)
        const float w0 = p00 * a02 + p01 * a12;
        const float w1 = p01 * a02 + p11 * a12;
        const float w2 = p02 * a02 + p12 * a12;
        const float apa02 = p02 + w0;
        const float apa12 = p12 + w1;
        const float apa22 = p22 + 2.0f * w2 + a02 * w0 + a12 * w1;
        // symmetrized M^T K
        const float mtk00 = m00 * k00 + m10 * k10;
        const float mtk11 = m01 * k01 + m11 * k11;
        const float mtk22 = m02 * k02 + m12 * k12;
        const float mtk01 = 0.5f * ((m00 * k01 + m10 * k11) + (m01 * k00 + m11 * k10));
        const float mtk02 = 0.5f * ((m00 * k02 + m10 * k12) + (m02 * k00 + m12 * k10));
        const float mtk12 = 0.5f * ((m01 * k02 + m11 * k12) + (m02 * k01 + m12 * k11));
        // Pn = Q + APA - sym(MtK)
        p00 = q0 + p00   - mtk00;
        p01 =      p01   - mtk01;
        p02 =      apa02 - mtk02;
        p11 = q1 + p11   - mtk11;
        p12 =      apa12 - mtk12;
        p22 = q2 + apa22 - mtk22;
    }

    // u = clip(K0 @ [x,y,th] + [v,om])   (u_fb = -K0 @ e0, e0 = -[x,y,th])
    float u0 = k00 * x + k01 * y + k02 * th + v;
    float u1 = k10 * x + k11 * y + k12 * th + om;
    u0 = fminf(fmaxf(u0, -V_MAXC), V_MAXC);
    u1 = fminf(fmaxf(u1, -W_MAXC), W_MAXC);

    v2f uo;
    uo.x = u0;
    uo.y = u1;
    *(v2f*)(out + (size_t)row * 2) = uo;   // coalesced 8B store
}

extern "C" void kernel_launch(void* const* d_in, const int* in_sizes, int n_in,
                              void* d_out, int out_size, void* d_ws, size_t ws_size,
                              hipStream_t stream) {
    (void)in_sizes; (void)n_in; (void)out_size; (void)d_ws; (void)ws_size;
    const float* latent = (const float*)d_in[0];
    const float* W      = (const float*)d_in[1];
    const float* bias   = (const float*)d_in[2];
    const float* q_raw  = (const float*)d_in[3];
    const float* r_raw  = (const float*)d_in[4];
    const float* qf_raw = (const float*)d_in[5];
    float* out = (float*)d_out;

    dim3 grid(BATCH / ROWS_PER_BLOCK);   // 128 workgroups
    dim3 block(256);                     // 8 wave32 per WG
    hipLaunchKernelGGL(lqr_action_net_kernel, grid, block, 0, stream,
                       latent, W, bias, q_raw, r_raw, qf_raw, out);
}